// MeshGraphNet_53291954208991
// MI455X (gfx1250) — compile-verified
//
#include <hip/hip_runtime.h>

#define HD      128
#define NNODES  20000
#define NEDGES  300000
#define NLAYERS 8
#define NI_D    6
#define EI_D    3
#define NO_D    3

typedef __attribute__((ext_vector_type(2))) float v2f;
typedef __attribute__((ext_vector_type(4))) float v4f;
typedef __attribute__((ext_vector_type(8))) float v8f;

// ---------------------------------------------------------------------------
// fp32 WMMA: D(16x16) = A(16x4) * B(4x16) + C   (wave32)
// ---------------------------------------------------------------------------
__device__ __forceinline__ v8f wmma4(v2f a, v2f b, v8f c) {
    return __builtin_amdgcn_wmma_f32_16x16x4_f32(
        /*neg_a=*/false, a, /*neg_b=*/false, b,
        /*c_mod=*/(short)0, c, /*reuse_a=*/false, /*reuse_b=*/false);
}

// ---------------------------------------------------------------------------
// Weight repack: W[K][H] row-major -> Wp[K/2][H][2] (pair-interleaved), so a
// B fragment (two consecutive K rows, same column) is one aligned b64 load.
// ---------------------------------------------------------------------------
__global__ void __launch_bounds__(256)
repack_kernel(const float* __restrict__ W, float* __restrict__ Wp, int K) {
    const int idx = blockIdx.x * 256 + threadIdx.x;    // over K*H
    if (idx >= K * HD) return;
    const int k = idx / HD, n = idx % HD;
    Wp[(((size_t)(k >> 1)) * HD + n) * 2 + (k & 1)] = W[idx];
}

// ---------------------------------------------------------------------------
// Core: 32x32 output macro-tile per wave (2 row-tiles x 2 col-tiles, 4 accs).
// A: [32 x K] row-major in LDS.  Bp: packed [K/2][H][2] in global.
// Per K-step: 2 ds b64 A-frags + 2 global b64 B-frags feed 4 WMMAs.
// ---------------------------------------------------------------------------
__device__ __forceinline__ void tile_gemm4(const float* As, int ldA,
                                           const float* __restrict__ Bp, int K,
                                           int n0, int n1,
                                           v8f& c00, v8f& c01,
                                           v8f& c10, v8f& c11) {
    const int lane = threadIdx.x & 31;
    const int half = lane >> 4;   // 0: K=k0,k0+1  1: K=k0+2,k0+3
    const int mrow = lane & 15;
    for (int k0 = 0; k0 < K; k0 += 4) {
        const int p = (k0 >> 1) + half;              // K-pair index
        v2f a0, a1, b0, b1;
        const float* ap0 = As + mrow * ldA + k0 + 2 * half;
        const float* ap1 = ap0 + 16 * ldA;
        a0.x = ap0[0];  a0.y = ap0[1];
        a1.x = ap1[0];  a1.y = ap1[1];
        b0 = *(const v2f*)(Bp + ((size_t)p * HD + n0 + mrow) * 2);
        b1 = *(const v2f*)(Bp + ((size_t)p * HD + n1 + mrow) * 2);
        c00 = wmma4(a0, b0, c00);
        c01 = wmma4(a0, b1, c01);
        c10 = wmma4(a1, b0, c10);
        c11 = wmma4(a1, b1, c11);
    }
}

// bias (+ReLU) then store a 16x16 tile (row-tile offset rt) into LDS [32 x ldH]
__device__ __forceinline__ void store_tile_lds(float* Hs, int ldH, int rt, int n0,
                                               const float* __restrict__ bias,
                                               v8f acc, bool relu) {
    const int lane = threadIdx.x & 31;
    const int half = lane >> 4;
    const int mrow = lane & 15;
    const int n = n0 + mrow;
    const float bv = bias[n];
#pragma unroll
    for (int i = 0; i < 8; ++i) {
        float v = acc[i] + bv;
        if (relu) v = fmaxf(v, 0.0f);
        Hs[(rt + half * 8 + i) * ldH + n] = v;
    }
}

__device__ __forceinline__ void store_tile_global(float* __restrict__ Og, int ldO,
                                                  int m0, int rt, int n0,
                                                  const float* __restrict__ bias,
                                                  v8f acc) {
    const int lane = threadIdx.x & 31;
    const int half = lane >> 4;
    const int mrow = lane & 15;
    const int n = n0 + mrow;
    const float bv = bias[n];
#pragma unroll
    for (int i = 0; i < 8; ++i)
        Og[(size_t)(m0 + rt + half * 8 + i) * ldO + n] = acc[i] + bv;
}

// ---------------------------------------------------------------------------
// Encoder: out[M,H] = relu(in[M,Kin] @ w1 + b1) @ w2p + b2   (Kin = 6 or 3)
// First GEMM tiny (K<=6) -> VALU; second GEMM (128x128) -> WMMA.
// 128 threads/block, 32 rows/block.
// ---------------------------------------------------------------------------
__global__ void __launch_bounds__(128)
encoder_kernel(const float* __restrict__ in, int Kin,
               const float* __restrict__ w1, const float* __restrict__ b1,
               const float* __restrict__ w2p, const float* __restrict__ b2,
               float* __restrict__ out) {
    __shared__ float Ain[32 * 8];
    __shared__ float Hs[32 * HD];
    const int m0 = blockIdx.x * 32;
    const int tid = threadIdx.x;

    for (int f = tid; f < 32 * Kin; f += 128) {
        int r = f / Kin, c = f % Kin;
        Ain[r * 8 + c] = in[(size_t)(m0 + r) * Kin + c];
    }
    __syncthreads();

    {   // thread t computes column t of h1 for all 32 rows
        const int n = tid;
        float w[8];
        for (int k = 0; k < Kin; ++k) w[k] = w1[k * HD + n];
        const float bv = b1[n];
        for (int r = 0; r < 32; ++r) {
            float s = bv;
            for (int k = 0; k < Kin; ++k) s += Ain[r * 8 + k] * w[k];
            Hs[r * HD + n] = fmaxf(s, 0.0f);
        }
    }
    __syncthreads();

    const int wave = tid >> 5;
    const int n0 = wave * 32, n1 = n0 + 16;
    v8f c00 = {}, c01 = {}, c10 = {}, c11 = {};
    tile_gemm4(Hs, HD, w2p, HD, n0, n1, c00, c01, c10, c11);
    store_tile_global(out, HD, m0, 0,  n0, b2, c00);
    store_tile_global(out, HD, m0, 0,  n1, b2, c01);
    store_tile_global(out, HD, m0, 16, n0, b2, c10);
    store_tile_global(out, HD, m0, 16, n1, b2, c11);
}

// ---------------------------------------------------------------------------
// Edge layer: e_new = relu([x[row]|x[col]|e] @ w1 + b1) @ w2 + b2 ; e <- e_new
//             agg[row] += e_new   (global fp32 atomics; agg pre-zeroed)
// 128 threads/block, 32 edges/block. A tile staged in LDS (32x384 = 48 KB).
// ---------------------------------------------------------------------------
__device__ __forceinline__ void emit_edge_tile(float* __restrict__ e,
                                               float* __restrict__ agg,
                                               const int* rs, int m0, int rt,
                                               int n0,
                                               const float* __restrict__ b2,
                                               v8f acc) {
    const int lane = threadIdx.x & 31;
    const int half = lane >> 4;
    const int mrow = lane & 15;
    const int n = n0 + mrow;
    const float bv = b2[n];
#pragma unroll
    for (int i = 0; i < 8; ++i) {
        const int ml = rt + half * 8 + i;
        const float v = acc[i] + bv;
        e[(size_t)(m0 + ml) * HD + n] = v;
        atomicAdd(&agg[(size_t)rs[ml] * HD + n], v);
    }
}

__global__ void __launch_bounds__(128)
edge_layer_kernel(const float* __restrict__ x,
                  float* __restrict__ e,
                  float* __restrict__ agg,
                  const int* __restrict__ rowi, const int* __restrict__ coli,
                  const float* __restrict__ w1p, const float* __restrict__ b1,
                  const float* __restrict__ w2p, const float* __restrict__ b2) {
    __shared__ float As[32 * (3 * HD)];   // 48 KB
    __shared__ float Hs[32 * HD];         // 16 KB
    __shared__ int rs[32];
    __shared__ int cs[32];
    const int m0 = blockIdx.x * 32;
    const int tid = threadIdx.x;

    if (tid < 32) rs[tid] = rowi[m0 + tid];
    else if (tid < 64) cs[tid - 32] = coli[m0 + tid - 32];
    __syncthreads();

    // stage [x[row] | x[col] | e] : 32 rows x 96 float4
    for (int f = tid; f < 32 * 96; f += 128) {
        const int r = f / 96, c4 = f % 96;
        v4f v;
        if (c4 < 32)
            v = *(const v4f*)(x + (size_t)rs[r] * HD + c4 * 4);
        else if (c4 < 64)
            v = *(const v4f*)(x + (size_t)cs[r] * HD + (c4 - 32) * 4);
        else
            v = *(const v4f*)(e + (size_t)(m0 + r) * HD + (c4 - 64) * 4);
        *(v4f*)(As + r * (3 * HD) + c4 * 4) = v;
    }
    __syncthreads();

    const int wave = tid >> 5;
    const int n0 = wave * 32, n1 = n0 + 16;
    v8f c00 = {}, c01 = {}, c10 = {}, c11 = {};
    tile_gemm4(As, 3 * HD, w1p, 3 * HD, n0, n1, c00, c01, c10, c11);  // 96 K-steps
    store_tile_lds(Hs, HD, 0,  n0, b1, c00, true);
    store_tile_lds(Hs, HD, 0,  n1, b1, c01, true);
    store_tile_lds(Hs, HD, 16, n0, b1, c10, true);
    store_tile_lds(Hs, HD, 16, n1, b1, c11, true);
    __syncthreads();

    v8f d00 = {}, d01 = {}, d10 = {}, d11 = {};
    tile_gemm4(Hs, HD, w2p, HD, n0, n1, d00, d01, d10, d11);          // 32 K-steps
    emit_edge_tile(e, agg, rs, m0, 0,  n0, b2, d00);
    emit_edge_tile(e, agg, rs, m0, 0,  n1, b2, d01);
    emit_edge_tile(e, agg, rs, m0, 16, n0, b2, d10);
    emit_edge_tile(e, agg, rs, m0, 16, n1, b2, d11);
}

// ---------------------------------------------------------------------------
// Node layer: x <- relu([x|agg] @ w1 + b1) @ w2 + b2 + x   (residual)
// ---------------------------------------------------------------------------
__device__ __forceinline__ void emit_node_tile(float* __restrict__ x,
                                               const float* As, int m0, int rt,
                                               int n0,
                                               const float* __restrict__ b2,
                                               v8f acc) {
    const int lane = threadIdx.x & 31;
    const int half = lane >> 4;
    const int mrow = lane & 15;
    const int n = n0 + mrow;
    const float bv = b2[n];
#pragma unroll
    for (int i = 0; i < 8; ++i) {
        const int ml = rt + half * 8 + i;
        x[(size_t)(m0 + ml) * HD + n] = acc[i] + bv + As[ml * (2 * HD) + n];
    }
}

__global__ void __launch_bounds__(128)
node_layer_kernel(float* __restrict__ x,
                  const float* __restrict__ agg,
                  const float* __restrict__ w1p, const float* __restrict__ b1,
                  const float* __restrict__ w2p, const float* __restrict__ b2) {
    __shared__ float As[32 * (2 * HD)];   // 32 KB  (first half is old x)
    __shared__ float Hs[32 * HD];         // 16 KB
    const int m0 = blockIdx.x * 32;
    const int tid = threadIdx.x;

    for (int f = tid; f < 32 * 64; f += 128) {
        const int r = f / 64, c4 = f % 64;
        v4f v;
        if (c4 < 32)
            v = *(const v4f*)(x + (size_t)(m0 + r) * HD + c4 * 4);
        else
            v = *(const v4f*)(agg + (size_t)(m0 + r) * HD + (c4 - 32) * 4);
        *(v4f*)(As + r * (2 * HD) + c4 * 4) = v;
    }
    __syncthreads();

    const int wave = tid >> 5;
    const int n0 = wave * 32, n1 = n0 + 16;
    v8f c00 = {}, c01 = {}, c10 = {}, c11 = {};
    tile_gemm4(As, 2 * HD, w1p, 2 * HD, n0, n1, c00, c01, c10, c11);  // 64 K-steps
    store_tile_lds(Hs, HD, 0,  n0, b1, c00, true);
    store_tile_lds(Hs, HD, 0,  n1, b1, c01, true);
    store_tile_lds(Hs, HD, 16, n0, b1, c10, true);
    store_tile_lds(Hs, HD, 16, n1, b1, c11, true);
    __syncthreads();

    v8f d00 = {}, d01 = {}, d10 = {}, d11 = {};
    tile_gemm4(Hs, HD, w2p, HD, n0, n1, d00, d01, d10, d11);
    emit_node_tile(x, As, m0, 0,  n0, b2, d00);
    emit_node_tile(x, As, m0, 0,  n1, b2, d01);
    emit_node_tile(x, As, m0, 16, n0, b2, d10);
    emit_node_tile(x, As, m0, 16, n1, b2, d11);
}

// ---------------------------------------------------------------------------
// Decoder: out[M,3] = relu(x @ w1 + b1) @ w2 + b2
// GEMM1 -> WMMA; 128->3 tail -> VALU.
// ---------------------------------------------------------------------------
__global__ void __launch_bounds__(128)
decoder_kernel(const float* __restrict__ x,
               const float* __restrict__ w1p, const float* __restrict__ b1,
               const float* __restrict__ w2, const float* __restrict__ b2,
               float* __restrict__ out) {
    __shared__ float As[32 * HD];
    __shared__ float Hs[32 * HD];
    const int m0 = blockIdx.x * 32;
    const int tid = threadIdx.x;

    for (int f = tid; f < 32 * 32; f += 128) {
        const int r = f / 32, c4 = f % 32;
        *(v4f*)(As + r * HD + c4 * 4) =
            *(const v4f*)(x + (size_t)(m0 + r) * HD + c4 * 4);
    }
    __syncthreads();

    const int wave = tid >> 5;
    const int n0 = wave * 32, n1 = n0 + 16;
    v8f c00 = {}, c01 = {}, c10 = {}, c11 = {};
    tile_gemm4(As, HD, w1p, HD, n0, n1, c00, c01, c10, c11);
    store_tile_lds(Hs, HD, 0,  n0, b1, c00, true);
    store_tile_lds(Hs, HD, 0,  n1, b1, c01, true);
    store_tile_lds(Hs, HD, 16, n0, b1, c10, true);
    store_tile_lds(Hs, HD, 16, n1, b1, c11, true);
    __syncthreads();

    if (tid < 32 * NO_D) {
        const int r = tid / NO_D, j = tid % NO_D;
        float s = b2[j];
        for (int k = 0; k < HD; ++k) s += Hs[r * HD + k] * w2[k * NO_D + j];
        out[(size_t)(m0 + r) * NO_D + j] = s;
    }
}

// ---------------------------------------------------------------------------
extern "C" void kernel_launch(void* const* d_in, const int* in_sizes, int n_in,
                              void* d_out, int out_size, void* d_ws, size_t ws_size,
                              hipStream_t stream) {
    const float* x_in    = (const float*)d_in[0];
    const float* e_in    = (const float*)d_in[1];
    const float* ne_w1   = (const float*)d_in[2];
    const float* ne_b1   = (const float*)d_in[3];
    const float* ne_w2   = (const float*)d_in[4];
    const float* ne_b2   = (const float*)d_in[5];
    const float* ee_w1   = (const float*)d_in[6];
    const float* ee_b1   = (const float*)d_in[7];
    const float* ee_w2   = (const float*)d_in[8];
    const float* ee_b2   = (const float*)d_in[9];
    const float* edge_w1 = (const float*)d_in[10];
    const float* edge_b1 = (const float*)d_in[11];
    const float* edge_w2 = (const float*)d_in[12];
    const float* edge_b2 = (const float*)d_in[13];
    const float* node_w1 = (const float*)d_in[14];
    const float* node_b1 = (const float*)d_in[15];
    const float* node_w2 = (const float*)d_in[16];
    const float* node_b2 = (const float*)d_in[17];
    const float* nd_w1   = (const float*)d_in[18];
    const float* nd_b1   = (const float*)d_in[19];
    const float* nd_w2   = (const float*)d_in[20];
    const float* nd_b2   = (const float*)d_in[21];
    const int*   eidx    = (const int*)d_in[22];
    const int*   rowi    = eidx;            // edge_index[0]
    const int*   coli    = eidx + NEDGES;   // edge_index[1]

    // ---- workspace layout (fp32) ----
    float* x   = (float*)d_ws;                       // [N, H]
    float* e   = x + (size_t)NNODES * HD;            // [E, H]
    float* agg = e + (size_t)NEDGES * HD;            // [N, H]
    float* pw  = agg + (size_t)NNODES * HD;          // packed weights
    float* p_ne_w2   = pw;                           // H*H
    float* p_ee_w2   = p_ne_w2   + (size_t)HD * HD;
    float* p_nd_w1   = p_ee_w2   + (size_t)HD * HD;
    float* p_edge_w1 = p_nd_w1   + (size_t)HD * HD;          // L*3H*H
    float* p_edge_w2 = p_edge_w1 + (size_t)NLAYERS * 3 * HD * HD;
    float* p_node_w1 = p_edge_w2 + (size_t)NLAYERS * HD * HD; // L*2H*H
    float* p_node_w2 = p_node_w1 + (size_t)NLAYERS * 2 * HD * HD;

    const dim3 blk(128);
    const dim3 rblk(256);
    const int gHH = (HD * HD + 255) / 256;

    // ---- repack all WMMA-consumed weights into pair-interleaved layout ----
    repack_kernel<<<gHH, rblk, 0, stream>>>(ne_w2, p_ne_w2, HD);
    repack_kernel<<<gHH, rblk, 0, stream>>>(ee_w2, p_ee_w2, HD);
    repack_kernel<<<gHH, rblk, 0, stream>>>(nd_w1, p_nd_w1, HD);
    for (int l = 0; l < NLAYERS; ++l) {
        repack_kernel<<<3 * gHH, rblk, 0, stream>>>(
            edge_w1 + (size_t)l * 3 * HD * HD, p_edge_w1 + (size_t)l * 3 * HD * HD, 3 * HD);
        repack_kernel<<<gHH, rblk, 0, stream>>>(
            edge_w2 + (size_t)l * HD * HD, p_edge_w2 + (size_t)l * HD * HD, HD);
        repack_kernel<<<2 * gHH, rblk, 0, stream>>>(
            node_w1 + (size_t)l * 2 * HD * HD, p_node_w1 + (size_t)l * 2 * HD * HD, 2 * HD);
        repack_kernel<<<gHH, rblk, 0, stream>>>(
            node_w2 + (size_t)l * HD * HD, p_node_w2 + (size_t)l * HD * HD, HD);
    }

    // ---- encoders ----
    encoder_kernel<<<NNODES / 32, blk, 0, stream>>>(x_in, NI_D, ne_w1, ne_b1,
                                                    p_ne_w2, ne_b2, x);
    encoder_kernel<<<NEDGES / 32, blk, 0, stream>>>(e_in, EI_D, ee_w1, ee_b1,
                                                    p_ee_w2, ee_b2, e);

    // ---- processor ----
    for (int l = 0; l < NLAYERS; ++l) {
        hipMemsetAsync(agg, 0, (size_t)NNODES * HD * sizeof(float), stream);
        edge_layer_kernel<<<NEDGES / 32, blk, 0, stream>>>(
            x, e, agg, rowi, coli,
            p_edge_w1 + (size_t)l * 3 * HD * HD, edge_b1 + (size_t)l * HD,
            p_edge_w2 + (size_t)l * HD * HD,     edge_b2 + (size_t)l * HD);
        node_layer_kernel<<<NNODES / 32, blk, 0, stream>>>(
            x, agg,
            p_node_w1 + (size_t)l * 2 * HD * HD, node_b1 + (size_t)l * HD,
            p_node_w2 + (size_t)l * HD * HD,     node_b2 + (size_t)l * HD);
    }

    // ---- decoder ----
    decoder_kernel<<<NNODES / 32, blk, 0, stream>>>(x, p_nd_w1, nd_b1, nd_w2,
                                                    nd_b2, (float*)d_out);
}